// Pool_85452669321471
// MI455X (gfx1250) — compile-verified
//
#include <hip/hip_runtime.h>
#include <hip/hip_bf16.h>

typedef __attribute__((ext_vector_type(2))) float v2f;
typedef __attribute__((ext_vector_type(8))) float v8f;

// ---------------------------------------------------------------------------
// Kernel 1: scores = sigmoid(h @ w + b), via V_WMMA_F32_16X16X4_F32.
// One wave (32 threads) computes 16 rows. A = h tile [16x4], B = w broadcast
// [4x16] (every column identical), C accumulates over D=128 in K-steps of 4.
// Column 0 of C is the score vector for the 16 rows.
// ---------------------------------------------------------------------------
__global__ __launch_bounds__(32)
void pool_scores_wmma(const float* __restrict__ h, const float* __restrict__ w,
                      const float* __restrict__ bptr, float* __restrict__ scores,
                      int N, int D) {
    const int b    = blockIdx.y;
    const int row0 = blockIdx.x * 16;
    const int lane = threadIdx.x;
    const int half = lane >> 4;   // 0: K=0,1   1: K=2,3
    const int m    = lane & 15;   // row within tile
    const int kofs = half * 2;

    const float* hrow = h + ((size_t)b * N + (row0 + m)) * D;

    v8f c = {};
    for (int k = 0; k < D; k += 4) {
        v2f a  = *(const v2f*)(hrow + k + kofs);   // A[m, k+kofs .. k+kofs+1]
        v2f bb = *(const v2f*)(w    + k + kofs);   // B[k+kofs.., n] (all n equal)
        c = __builtin_amdgcn_wmma_f32_16x16x4_f32(
                /*neg_a=*/false, a, /*neg_b=*/false, bb,
                /*c_mod=*/(short)0, c, /*reuse_a=*/false, /*reuse_b=*/false);
    }

    // C layout: lane 0 holds M=0..7 (N=0) in c[0..7]; lane 16 holds M=8..15.
    if (m == 0) {
        const float bias = *bptr;
        float* srow = scores + (size_t)b * N + row0 + half * 8;
        #pragma unroll
        for (int v = 0; v < 8; ++v) {
            float x = c[v] + bias;
            srow[v] = 1.0f / (1.0f + __expf(-x));
        }
    }
}

// ---------------------------------------------------------------------------
// Kernel 2: top-K selection per batch. Full LDS bitonic sort of 4096 packed
// keys (orderable-score << 32 | (N-1-i)) sorted descending => top-K in order,
// ties broken by smaller index first (matches jax.lax.top_k).
// ---------------------------------------------------------------------------
#define SORT_N 4096
__global__ __launch_bounds__(1024)
void pool_topk_sort(const float* __restrict__ scores, int* __restrict__ idx_out,
                    float* __restrict__ vals_out, float* __restrict__ idx_f_out,
                    int N, int K) {
    __shared__ unsigned long long keys[SORT_N];
    const int b   = blockIdx.x;
    const int tid = threadIdx.x;
    const float* s = scores + (size_t)b * N;

    for (int e = tid; e < N; e += blockDim.x) {
        unsigned int bits = __float_as_uint(s[e]);
        // map float -> monotonically increasing unsigned
        unsigned int u = bits ^ ((bits & 0x80000000u) ? 0xFFFFFFFFu : 0x80000000u);
        keys[e] = ((unsigned long long)u << 32) | (unsigned int)(N - 1 - e);
    }
    __syncthreads();

    for (int k = 2; k <= N; k <<= 1) {
        for (int j = k >> 1; j > 0; j >>= 1) {
            for (int e = tid; e < N; e += blockDim.x) {
                int p = e ^ j;
                if (p > e) {
                    bool desc = ((e & k) == 0);
                    unsigned long long a = keys[e], c = keys[p];
                    if (desc ? (a < c) : (a > c)) { keys[e] = c; keys[p] = a; }
                }
            }
            __syncthreads();
        }
    }

    for (int p = tid; p < K; p += blockDim.x) {
        unsigned long long kk = keys[p];
        int i = (N - 1) - (int)(kk & 0xFFFFFFFFu);
        unsigned int u = (unsigned int)(kk >> 32);
        unsigned int bits = u ^ ((u & 0x80000000u) ? 0x80000000u : 0xFFFFFFFFu);
        idx_out[(size_t)b * K + p]   = i;
        vals_out[(size_t)b * K + p]  = __uint_as_float(bits);
        idx_f_out[(size_t)b * K + p] = (float)i;   // idx region of d_out (float dtype)
    }
}

// ---------------------------------------------------------------------------
// Kernel 3: new_h[b,i,:] = h[b, idx[b,i], :] * vals[b,i]   (coalesced)
// ---------------------------------------------------------------------------
__global__ __launch_bounds__(128)
void pool_gather_h(const float* __restrict__ h, const int* __restrict__ idx,
                   const float* __restrict__ vals, float* __restrict__ out_h,
                   int N, int D, int K) {
    const int i = blockIdx.x, b = blockIdx.y, t = threadIdx.x;
    const int   r = idx[(size_t)b * K + i];
    const float v = vals[(size_t)b * K + i];
    const float* src = h + ((size_t)b * N + r) * D;
    float* dst = out_h + ((size_t)b * K + i) * D;
    dst[t] = src[t] * v;
}

// ---------------------------------------------------------------------------
// Kernel 4: g_new[b,i,j] = g[b, idx[i], idx[j]] / (deg + eps).
// One block per output row. Since K = N/2, virtually every cacheline of the
// source row is needed anyway, so stream the WHOLE 16KB row into LDS with
// coalesced b128 loads, then do both gather passes (degree sum + normalized
// b128 write) from LDS. 25KB LDS/block -> ~12 blocks per WGP.
// ---------------------------------------------------------------------------
#define POOL_N 4096
#define POOL_K 2048
__global__ __launch_bounds__(256)
void pool_gather_norm_g(const float* __restrict__ g, const int* __restrict__ idx,
                        float* __restrict__ out_g, int N, int K) {
    __shared__ float row[POOL_N];     // 16 KB: full source row
    __shared__ int   idx_lds[POOL_K]; //  8 KB: selected column indices
    __shared__ float red[256];        //  1 KB: reduction scratch

    const int i = blockIdx.x, b = blockIdx.y, tid = threadIdx.x;
    const int r = idx[(size_t)b * K + i];
    const float* grow = g + ((size_t)b * N + r) * N;

    // Stage the full row into LDS, 16B per thread per iteration (b128 loads).
    for (int j = tid * 4; j < N; j += 256 * 4) {
        float4 v4 = *(const float4*)(grow + j);
        row[j + 0] = v4.x; row[j + 1] = v4.y;
        row[j + 2] = v4.z; row[j + 3] = v4.w;
    }
    for (int j = tid; j < K; j += 256) idx_lds[j] = idx[(size_t)b * K + j];
    __syncthreads();

    // Pass 1: degree = sum over selected columns (gather from LDS).
    float local = 0.0f;
    for (int j = tid; j < K; j += 256) local += row[idx_lds[j]];
    red[tid] = local;
    __syncthreads();
    #pragma unroll
    for (int s = 128; s > 0; s >>= 1) {
        if (tid < s) red[tid] += red[tid + s];
        __syncthreads();
    }
    const float scale = 1.0f / (red[0] + 1e-10f);

    // Pass 2: normalized gather, packed into b128 stores.
    float* dst = out_g + ((size_t)b * K + i) * K;
    for (int j0 = tid * 4; j0 < K; j0 += 256 * 4) {
        float4 o;
        o.x = row[idx_lds[j0 + 0]] * scale;
        o.y = row[idx_lds[j0 + 1]] * scale;
        o.z = row[idx_lds[j0 + 2]] * scale;
        o.w = row[idx_lds[j0 + 3]] * scale;
        *(float4*)(dst + j0) = o;
    }
}

// ---------------------------------------------------------------------------
extern "C" void kernel_launch(void* const* d_in, const int* in_sizes, int n_in,
                              void* d_out, int out_size, void* d_ws, size_t ws_size,
                              hipStream_t stream) {
    const float* g = (const float*)d_in[0];   // [B,N,N]
    const float* h = (const float*)d_in[1];   // [B,N,D]
    const float* w = (const float*)d_in[2];   // [D]
    const float* b = (const float*)d_in[3];   // scalar

    const int D = in_sizes[2];                         // 128
    const int N = (int)((long long)in_sizes[0] / in_sizes[1] * D); // 4096
    const int B = (int)(in_sizes[1] / ((long long)N * D));         // 4
    const int K = (N / 2 < 2) ? 2 : N / 2;             // 2048

    // workspace: scores [B*N] | idx [B*K] (int) | vals [B*K]
    float* ws_scores = (float*)d_ws;
    int*   ws_idx    = (int*)(ws_scores + (size_t)B * N);
    float* ws_vals   = (float*)(ws_idx + (size_t)B * K);

    // output: g_new [B*K*K] | new_h [B*K*D] | idx [B*K] (as float)
    float* out_g   = (float*)d_out;
    float* out_h   = out_g + (size_t)B * K * K;
    float* out_idx = out_h + (size_t)B * K * D;

    pool_scores_wmma<<<dim3(N / 16, B), 32, 0, stream>>>(h, w, b, ws_scores, N, D);
    pool_topk_sort<<<B, 1024, 0, stream>>>(ws_scores, ws_idx, ws_vals, out_idx, N, K);
    pool_gather_h<<<dim3(K, B), D, 0, stream>>>(h, ws_idx, ws_vals, out_h, N, D, K);
    pool_gather_norm_g<<<dim3(K, B), 256, 0, stream>>>(g, ws_idx, out_g, N, K);
}